// ConvGen154_65481071395520
// MI455X (gfx1250) — compile-verified
//
#include <hip/hip_runtime.h>

// out[b,0,i,j] = sigmoid( cos(pi*x[b,0,i,j] + w0) - thr ),  (i,j) in [0,510)^2
// Memory-bound streaming map (133 MB @ 23.3 TB/s => ~5.7 us floor).
// Data path: global_load_b128 (NT) -> v_cos/v_exp/v_rcp (trans pipe) ->
// 2x global_store_b64 (NT; output rows are only 8B aligned), plus an explicit
// global_prefetch_b8 of the block's next row (8 MB ahead, lands in 192MB L2).

namespace {
constexpr int H  = 512, W  = 512;
constexpr int OH = 510, OW = 510;
constexpr int B  = 64;
constexpr int GRID = 4080;                    // 8 * 510 => row index i invariant per block
constexpr float INV_2PI = 0.15915494309189535f;
constexpr float LOG2E   = 1.4426950408889634f;
}

typedef __attribute__((ext_vector_type(4))) float v4f;
typedef __attribute__((ext_vector_type(2))) float v2f;

// v_cos_f32 takes revolutions: cos(pi*x + w0) == v_cos(0.5*x + w0/(2*pi)).
// sigmoid(t) = 1 / (1 + exp2(-t * log2(e)))  -> v_exp_f32 + v_rcp_f32.
__device__ __forceinline__ float crop_op(float xv, float w_rev, float th) {
    float rev = __builtin_fmaf(xv, 0.5f, w_rev);
    float c   = __builtin_amdgcn_cosf(rev);            // v_cos_f32
    float t   = c - th;
    float e   = __builtin_amdgcn_exp2f(-t * LOG2E);    // v_exp_f32
    return __builtin_amdgcn_rcpf(1.0f + e);            // v_rcp_f32
}

__global__ __launch_bounds__(128) void crop_cos_sigmoid_rows(
    const float* __restrict__ x,     // (64,1,512,512)
    const float* __restrict__ w,     // (9,) -- only w[0] used
    const float* __restrict__ thr,   // (1,)
    float* __restrict__ out)         // (64,1,510,510)
{
    const float w_rev = w[0] * INV_2PI;   // uniform -> scalar loads
    const float th    = thr[0];
    const int   t     = threadIdx.x;

    // Block handles rows r0, r0+4080, ..., r0+7*4080. Since 4080 = 8*510,
    // the within-image row i is constant; only the image index steps by 8.
    const int r0 = blockIdx.x;            // < 4080
    const int bi0 = r0 / OH;              // 0..7 (one division, at entry)
    const int i   = r0 - bi0 * OH;

    const float* __restrict__ src = x   + (size_t)bi0 * (H * W) + (size_t)i * W; // 16B aligned
    float*       __restrict__ dst = out + (size_t)r0 * OW;                       // 8B aligned

    constexpr size_t SRC_STEP = (size_t)8 * H * W;    // 8 images
    constexpr size_t DST_STEP = (size_t)8 * OH * OW;  // 4080 rows

    for (int bi = bi0; bi < B; bi += 8) {
        // Prefetch the row this block processes next (8 images = 8 MB ahead).
        // 8 lanes x 256B covers the 2048B input row.
        if (bi + 8 < B && t < 8) {
            unsigned long long pa =
                (unsigned long long)(const void*)((const char*)(src + SRC_STEP) + t * 256);
            asm volatile("global_prefetch_b8 %0, off" :: "v"(pa) : "memory");
        }

        if (t < 127) {
            // 127 * 4 = 508 elements via aligned float4 loads
            const v4f xv = __builtin_nontemporal_load((const v4f*)src + t);  // global_load_b128 NT
            v2f lo, hi;
            lo.x = crop_op(xv.x, w_rev, th);
            lo.y = crop_op(xv.y, w_rev, th);
            hi.x = crop_op(xv.z, w_rev, th);
            hi.y = crop_op(xv.w, w_rev, th);
            v2f* d = (v2f*)(dst + 4 * t);              // 8B-aligned always (510*4 % 8 == 0)
            __builtin_nontemporal_store(lo, d);        // global_store_b64 NT
            __builtin_nontemporal_store(hi, d + 1);
        } else {
            // tail: elements 508, 509 (compiler merges to b64 ld/st)
            float a0 = __builtin_nontemporal_load(src + 508);
            float a1 = __builtin_nontemporal_load(src + 509);
            __builtin_nontemporal_store(crop_op(a0, w_rev, th), dst + 508);
            __builtin_nontemporal_store(crop_op(a1, w_rev, th), dst + 509);
        }

        src += SRC_STEP;
        dst += DST_STEP;
    }
}

extern "C" void kernel_launch(void* const* d_in, const int* in_sizes, int n_in,
                              void* d_out, int out_size, void* d_ws, size_t ws_size,
                              hipStream_t stream) {
    (void)in_sizes; (void)n_in; (void)out_size; (void)d_ws; (void)ws_size;
    const float* x   = (const float*)d_in[0];
    const float* w   = (const float*)d_in[1];
    const float* thr = (const float*)d_in[2];
    float* out = (float*)d_out;

    // 4080 blocks x 8 row-iterations = 32640 rows; 4 waves/block -> 16320 waves.
    dim3 grid(GRID), block(128);
    hipLaunchKernelGGL(crop_cos_sigmoid_rows, grid, block, 0, stream, x, w, thr, out);
}